// Transformer_60172491817907
// MI455X (gfx1250) — compile-verified
//
#include <hip/hip_runtime.h>
#include <hip/hip_bf16.h>
#include <math.h>

// ---------------------------------------------------------------------------
// MI455X (gfx1250) implementation. FP32 WMMA 16x16x4 for all GEMMs (op is
// memory-bound: ~14.5 GFLOP vs ~200MB @ 23.3TB/s, so fp32 keeps reference
// precision at no cost). LDS staging uses the CDNA5 async global->LDS path.
// ---------------------------------------------------------------------------

typedef __attribute__((ext_vector_type(2))) float v2f;
typedef __attribute__((ext_vector_type(4))) float v4f;
typedef __attribute__((ext_vector_type(8))) float v8f;
typedef __attribute__((ext_vector_type(4))) int   v4i;

#define AS_GLOBAL __attribute__((address_space(1)))
#define AS_LDS    __attribute__((address_space(3)))

#if defined(__AMDGCN__) && __has_builtin(__builtin_amdgcn_global_load_async_to_lds_b128)
#define HAVE_ASYNC_LDS 1
#else
#define HAVE_ASYNC_LDS 0
#endif

__device__ __forceinline__ void cp16_g2l(const float* g, float* l) {
#if HAVE_ASYNC_LDS
    __builtin_amdgcn_global_load_async_to_lds_b128(
        (AS_GLOBAL v4i*)g, (AS_LDS v4i*)l, 0, 0);
#else
    *(v4f*)l = *(const v4f*)g;
#endif
}

__device__ __forceinline__ void async_copies_wait() {
#if HAVE_ASYNC_LDS
#if __has_builtin(__builtin_amdgcn_s_wait_asynccnt)
    __builtin_amdgcn_s_wait_asynccnt(0);
#else
    asm volatile("s_wait_asynccnt 0" ::: "memory");
#endif
#endif
}

// ---------------------------------------------------------------------------
// Generic WMMA GEMM: out[m,n] = act( sum_k A[m,k]*W[n,k] + bias[n] (+res[m,n]) )
// A: [M x K] row-major, W: [N x K] row-major (i.e. out = A @ W^T).
// Block = 128 threads (4 waves). Block tile: 16 (M) x 64 (N). K chunked by 64.
// M % 16 == 0, K % 64 == 0. N arbitrary (guarded).
// ---------------------------------------------------------------------------
#define GEMM_BK 64

template<bool RELU, bool RES>
__global__ __launch_bounds__(128)
void wmma_gemm_kernel(const float* __restrict__ A, const float* __restrict__ W,
                      const float* __restrict__ bias, const float* __restrict__ res,
                      float* __restrict__ out, int M, int N, int K)
{
    __shared__ __align__(16) float lA[16 * GEMM_BK];
    __shared__ __align__(16) float lB[64 * GEMM_BK];

    const int tid  = threadIdx.x;
    const int wave = tid >> 5;
    const int lane = tid & 31;
    const int l16  = lane & 15;
    const int half = lane >> 4;          // 0: K 0,1  /  1: K 2,3 within a step

    const int m0   = blockIdx.y * 16;
    const int nblk = blockIdx.x * 64;

    v8f acc = {};

    for (int k0 = 0; k0 < K; k0 += GEMM_BK) {
        // ---- stage A tile: 16 rows x 64 floats (256 x b128) ----
        for (int i = tid; i < 16 * (GEMM_BK / 4); i += 128) {
            int r  = i >> 4;
            int c4 = (i & 15) * 4;
            cp16_g2l(&A[(size_t)(m0 + r) * K + k0 + c4], &lA[r * GEMM_BK + c4]);
        }
        // ---- stage B tile: 64 rows x 64 floats (1024 x b128), row-guarded ----
        for (int i = tid; i < 64 * (GEMM_BK / 4); i += 128) {
            int r  = i >> 4;
            int c4 = (i & 15) * 4;
            int n  = nblk + r;
            if (n < N) {
                cp16_g2l(&W[(size_t)n * K + k0 + c4], &lB[r * GEMM_BK + c4]);
            } else {
                v4f z = {};
                *(v4f*)&lB[r * GEMM_BK + c4] = z;
            }
        }
        async_copies_wait();
        __syncthreads();

        // ---- 16 WMMA steps over this K chunk ----
        const float* aRow = &lA[l16 * GEMM_BK + 2 * half];
        const float* bRow = &lB[(wave * 16 + l16) * GEMM_BK + 2 * half];
#if __has_builtin(__builtin_amdgcn_wmma_f32_16x16x4_f32)
#pragma unroll
        for (int kk = 0; kk < GEMM_BK; kk += 4) {
            v2f a = *(const v2f*)(aRow + kk);   // A[m=l16][k0+kk+2*half+{0,1}]
            v2f b = *(const v2f*)(bRow + kk);   // W[n][k0+kk+2*half+{0,1}]
            acc = __builtin_amdgcn_wmma_f32_16x16x4_f32(
                false, a, false, b, (short)0, acc, false, false);
        }
#else
        // scalar fallback (keeps kernel compiling if builtin missing)
        for (int kk = 0; kk < GEMM_BK; kk += 4)
            for (int r = 0; r < 8; ++r)
                for (int j = 0; j < 4; ++j)
                    acc[r] += lA[(r + 8 * half) * GEMM_BK + kk + j] *
                              lB[(wave * 16 + l16) * GEMM_BK + kk + j];
#endif
        __syncthreads();
    }

    // ---- epilogue: lane holds column n = nblk + wave*16 + l16, rows r+8*half ----
    const int n = nblk + wave * 16 + l16;
    if (n < N) {
        const float bv = bias ? bias[n] : 0.0f;
#pragma unroll
        for (int r = 0; r < 8; ++r) {
            int m = m0 + r + 8 * half;
            float x = acc[r] + bv;
            if (RES)  x += res[(size_t)m * N + n];
            if (RELU) x = fmaxf(x, 0.0f);
            out[(size_t)m * N + n] = x;
        }
    }
}

// ---------------------------------------------------------------------------
// pos-MLP layer 1: x1 = relu(ref_windows @ pw1^T + pb1), K = 8 (too small for WMMA)
// ---------------------------------------------------------------------------
__global__ void pos1_kernel(const float* __restrict__ rw, const float* __restrict__ pw1,
                            const float* __restrict__ pb1, float* __restrict__ x1)
{
    int idx = blockIdx.x * blockDim.x + threadIdx.x;   // Q*256 total
    int qi = idx >> 8, d = idx & 255;
    float s = pb1[d];
#pragma unroll
    for (int j = 0; j < 8; ++j) s += rw[qi * 8 + j] * pw1[d * 8 + j];
    x1[idx] = fmaxf(s, 0.0f);
}

// ---------------------------------------------------------------------------
// softmax over 25 points per (q, head).  a: [Q x 200] -> aw: [Q x 200]
// ---------------------------------------------------------------------------
__global__ void softmax25_kernel(const float* __restrict__ a, float* __restrict__ aw)
{
    int idx = blockIdx.x * blockDim.x + threadIdx.x;   // Q*8 rows
    const float* row = a  + (size_t)(idx >> 3) * 200 + (idx & 7) * 25;
    float*       dst = aw + (size_t)(idx >> 3) * 200 + (idx & 7) * 25;
    float mx = row[0];
    for (int p = 1; p < 25; ++p) mx = fmaxf(mx, row[p]);
    float s = 0.0f;
    float e[25];
    for (int p = 0; p < 25; ++p) { e[p] = __expf(row[p] - mx); s += e[p]; }
    float inv = 1.0f / s;
    for (int p = 0; p < 25; ++p) dst[p] = e[p] * inv;
}

// ---------------------------------------------------------------------------
// deformable bilinear sampling + weighted sum over 25 points.
// block = query (4096 blocks x 256 threads); thread = (head, channel)
// v layout: [HW x 256], channel = h*32 + d.  attn: [Q x 256]
// ---------------------------------------------------------------------------
__global__ __launch_bounds__(256)
void deform_kernel(const float* __restrict__ rw, const float* __restrict__ aw,
                   const float* __restrict__ ob, const float* __restrict__ v,
                   float* __restrict__ attn)
{
    const int qi = blockIdx.x;
    const int h  = threadIdx.x >> 5;
    const int d  = threadIdx.x & 31;

    const float* r  = rw + (size_t)qi * 8;
    const float cx = r[0], cy = r[1], w0 = r[3], h0 = r[4], ang0 = r[6];
    const float* o5 = ob + (size_t)qi * 40 + h * 5;

    const float ang = (ang0 + o5[4] * (1.0f / 16.0f)) * 6.28318530717958647692f;
    const float bx  = cx + o5[0] * (1.0f / 8.0f) * w0;
    const float by  = cy + o5[1] * (1.0f / 8.0f) * h0;
    const float bw  = w0 + o5[2] * (1.0f / 8.0f) * w0;
    const float bh  = h0 + o5[3] * (1.0f / 8.0f) * h0;
    const float cs = cosf(ang), sn = sinf(ang);

    const float* aww = aw + (size_t)qi * 200 + h * 25;
    const float* vch = v + h * 32 + d;

    float acc = 0.0f;
    for (int p = 0; p < 25; ++p) {
        float kgx = (float)((p % 5) - 2) * 0.2f;
        float kgy = (float)((p / 5) - 2) * 0.2f;
        float gx0 = kgx * bw, gy0 = kgy * bh;
        float px = bx + gx0 * cs - gy0 * sn;
        float py = by + gx0 * sn + gy0 * cs;
        float xf = px * 256.0f - 0.5f;
        float yf = py * 256.0f - 0.5f;
        float x0 = floorf(xf), y0 = floorf(yf);
        float fx = xf - x0, fy = yf - y0;
        int xi = (int)x0, yi = (int)y0;

        float s = 0.0f;
#pragma unroll
        for (int c = 0; c < 4; ++c) {
            int xx = xi + (c & 1);
            int yy = yi + (c >> 1);
            float wgt = ((c & 1) ? fx : 1.0f - fx) * ((c >> 1) ? fy : 1.0f - fy);
            bool valid = (xx >= 0) && (xx <= 255) && (yy >= 0) && (yy <= 255);
            int xc = xx < 0 ? 0 : (xx > 255 ? 255 : xx);
            int yc = yy < 0 ? 0 : (yy > 255 ? 255 : yy);
            float val = vch[((size_t)yc * 256 + xc) * 256];
            s += valid ? val * wgt : 0.0f;
        }
        acc += aww[p] * s;
    }
    attn[(size_t)qi * 256 + h * 32 + d] = acc;
}

// ---------------------------------------------------------------------------
// out = LayerNorm(x + y) * g + b over D=256.  block = row, 256 threads.
// ---------------------------------------------------------------------------
__global__ __launch_bounds__(256)
void add_ln_kernel(const float* __restrict__ x, const float* __restrict__ y,
                   const float* __restrict__ g, const float* __restrict__ b,
                   float* __restrict__ out)
{
    const int qi = blockIdx.x, d = threadIdx.x;
    const size_t idx = (size_t)qi * 256 + d;
    float v = x[idx] + y[idx];

    __shared__ float red[8];
    float s = v;
#pragma unroll
    for (int o = 16; o; o >>= 1) s += __shfl_xor(s, o, 32);
    if ((d & 31) == 0) red[d >> 5] = s;
    __syncthreads();
    float tot = 0.0f;
#pragma unroll
    for (int i = 0; i < 8; ++i) tot += red[i];
    const float mean = tot * (1.0f / 256.0f);
    __syncthreads();

    const float dv = v - mean;
    s = dv * dv;
#pragma unroll
    for (int o = 16; o; o >>= 1) s += __shfl_xor(s, o, 32);
    if ((d & 31) == 0) red[d >> 5] = s;
    __syncthreads();
    tot = 0.0f;
#pragma unroll
    for (int i = 0; i < 8; ++i) tot += red[i];
    const float var = tot * (1.0f / 256.0f);

    out[idx] = dv * rsqrtf(var + 1e-5f) * g[d] + b[d];
}

// ---------------------------------------------------------------------------
// Host-side orchestration
// ---------------------------------------------------------------------------
static inline void launch_gemm(bool relu, bool res,
                               const float* A, const float* W, const float* bias,
                               const float* resp, float* out,
                               int M, int N, int K, hipStream_t stream)
{
    dim3 grid((N + 63) / 64, M / 16);
    dim3 block(128);
    if (relu)      hipLaunchKernelGGL((wmma_gemm_kernel<true,  false>), grid, block, 0, stream, A, W, bias, resp, out, M, N, K);
    else if (res)  hipLaunchKernelGGL((wmma_gemm_kernel<false, true >), grid, block, 0, stream, A, W, bias, resp, out, M, N, K);
    else           hipLaunchKernelGGL((wmma_gemm_kernel<false, false>), grid, block, 0, stream, A, W, bias, resp, out, M, N, K);
}

extern "C" void kernel_launch(void* const* d_in, const int* in_sizes, int n_in,
                              void* d_out, int out_size, void* d_ws, size_t ws_size,
                              hipStream_t stream)
{
    const float* query   = (const float*)d_in[0];   // [1,4096,256]
    const float* memory  = (const float*)d_in[1];   // [1,65536,256]
    const float* refw    = (const float*)d_in[2];   // [1,4096,8]
    // d_in[3] = memory_shape (int32, [256,256]) -- compile-time constants used
    const float* pw1     = (const float*)d_in[4];
    const float* pb1     = (const float*)d_in[5];
    const float* pw2     = (const float*)d_in[6];
    const float* pb2     = (const float*)d_in[7];
    const float* pw3     = (const float*)d_in[8];
    const float* pb3     = (const float*)d_in[9];
    const float* wa      = (const float*)d_in[10];
    const float* ba      = (const float*)d_in[11];
    const float* wb      = (const float*)d_in[12];
    const float* bb      = (const float*)d_in[13];
    const float* wv      = (const float*)d_in[14];
    const float* bv      = (const float*)d_in[15];
    const float* wo      = (const float*)d_in[16];
    const float* bo      = (const float*)d_in[17];
    const float* lin1_w  = (const float*)d_in[18];
    const float* lin1_b  = (const float*)d_in[19];
    const float* lin2_w  = (const float*)d_in[20];
    const float* lin2_b  = (const float*)d_in[21];
    const float* ln2_g   = (const float*)d_in[22];
    const float* ln2_b   = (const float*)d_in[23];
    const float* ln3_g   = (const float*)d_in[24];
    const float* ln3_b   = (const float*)d_in[25];
    float* outp = (float*)d_out;

    const int Q = 4096, D = 256, HW = 65536, FF = 1024;

    float* ws = (float*)d_ws;
    size_t off = 0;
    auto wsa = [&](size_t n) { float* p = ws + off; off += n; return p; };
    float* vbuf  = wsa((size_t)HW * D);   // 65536 x 256
    float* x1    = wsa((size_t)Q * D);
    float* x2    = wsa((size_t)Q * D);
    float* qbuf  = wsa((size_t)Q * D);
    float* araw  = wsa((size_t)Q * 200);
    float* awbuf = wsa((size_t)Q * 200);
    float* obbuf = wsa((size_t)Q * 40);
    float* attn  = wsa((size_t)Q * D);
    float* attn2 = wsa((size_t)Q * D);
    float* xln2  = wsa((size_t)Q * D);
    float* h1    = wsa((size_t)Q * FF);
    float* h2    = wsa((size_t)Q * D);
    (void)ws_size; (void)in_sizes; (void)n_in; (void)out_size;

    // 1) pos MLP
    hipLaunchKernelGGL(pos1_kernel, dim3(Q), dim3(256), 0, stream, refw, pw1, pb1, x1);
    launch_gemm(true,  false, x1, pw2, pb2, nullptr, x2,   Q, D, D, stream);   // relu
    launch_gemm(false, true,  x2, pw3, pb3, query,   qbuf, Q, D, D, stream);   // q = query + qpos

    // 2) v projection (big GEMM, 65536 x 256 x 256)
    launch_gemm(false, false, memory, wv, bv, nullptr, vbuf, HW, D, D, stream);

    // 3) attention weights & sampling offsets
    launch_gemm(false, false, qbuf, wa, ba, nullptr, araw,  Q, 200, D, stream);
    launch_gemm(false, false, qbuf, wb, bb, nullptr, obbuf, Q, 40,  D, stream);
    hipLaunchKernelGGL(softmax25_kernel, dim3(Q * 8 / 256), dim3(256), 0, stream, araw, awbuf);

    // 4) deformable bilinear sampling + point-weighted sum
    hipLaunchKernelGGL(deform_kernel, dim3(Q), dim3(256), 0, stream,
                       refw, awbuf, obbuf, vbuf, attn);

    // 5) output projection, residual + LN
    launch_gemm(false, false, attn, wo, bo, nullptr, attn2, Q, D, D, stream);
    hipLaunchKernelGGL(add_ln_kernel, dim3(Q), dim3(256), 0, stream,
                       query, attn2, ln2_g, ln2_b, xln2);

    // 6) FFN + residual + LN
    launch_gemm(true,  false, xln2, lin1_w, lin1_b, nullptr, h1, Q, FF, D,  stream);
    launch_gemm(false, false, h1,   lin2_w, lin2_b, nullptr, h2, Q, D,  FF, stream);
    hipLaunchKernelGGL(add_ln_kernel, dim3(Q), dim3(256), 0, stream,
                       xln2, h2, ln3_g, ln3_b, outp);
}